// ProjSpatialBlock_53944789238105
// MI455X (gfx1250) — compile-verified
//
#include <hip/hip_runtime.h>
#include <hip/hip_bf16.h>

// ---------------------------------------------------------------------------
// MI455X (gfx1250) GCNet-style global-context block.
// Heavy einsums: fp32 WMMA (V_WMMA_F32_16X16X4_F32), 64x64 workgroup tiles,
// double-buffered LDS fed by GLOBAL_LOAD_ASYNC_TO_LDS_B128 (ASYNCcnt).
// x (151 MB) is L2-resident (192 MB L2) -> second pass over x is L2 traffic.
// ---------------------------------------------------------------------------

typedef __attribute__((ext_vector_type(2))) float v2f;
typedef __attribute__((ext_vector_type(8))) float v8f;

#define C_DIM 512
#define N_DIM 9216   // 96*96
#define B_DIM 8
#define P_DIM 128
#define M_DIM 8
#define PL_DIM 32
#define RROWS 264    // 128 mproj + 128 vproj + 8 mask
#define RY    272    // Y rows per batch (RROWS padded to /16)
#define RPAD  320    // packed-weight rows padded (5 * 64)
#define KC    32     // K chunk
#define LDA_S 36     // LDS row stride (floats) for A / B[n][k]: 144B, 16B-aligned
#define LDB_KN 68    // LDS row stride (floats) for B[k][n]: 272B, 16B-aligned

#define USE_ASYNC 1

#define OFF_WCAT 0
#define OFF_BCAT 163840
#define OFF_Y    165888

// -------------------------- async helpers ----------------------------------
__device__ __forceinline__ uint32_t lds_off(const void* p) {
  // Generic LDS addresses carry the byte offset in addr[31:0] (ISA 10.2).
  return (uint32_t)(uintptr_t)p;
}

__device__ __forceinline__ void async_ld_b128(uint32_t ldsByteOff, const float* g) {
#if USE_ASYNC
  asm volatile("global_load_async_to_lds_b128 %0, %1, off"
               :: "v"(ldsByteOff), "v"(g) : "memory");
#else
  // fallback: synchronous copy
  float4 v = *(const float4*)g;
  *(float4*)((char*)nullptr + 0) = v;  // unused
#endif
}

__device__ __forceinline__ void async_wait0() {
#if USE_ASYNC
  asm volatile("s_wait_asynccnt 0x0" ::: "memory");
#endif
}

__device__ __forceinline__ v8f wmma_f32(v2f a, v2f b, v8f c) {
  return __builtin_amdgcn_wmma_f32_16x16x4_f32(false, a, false, b, (short)0, c,
                                               false, false);
}

// ---------------------------------------------------------------------------
// Pack [w_mproj; w_vproj; w_mask] -> Wcat[RPAD][C], biases -> bcat[RPAD].
// ---------------------------------------------------------------------------
__global__ void __launch_bounds__(256)
prep_kernel(const float* __restrict__ wm, const float* __restrict__ bm,
            const float* __restrict__ wmp, const float* __restrict__ bmp,
            const float* __restrict__ wvp, const float* __restrict__ bvp,
            float* __restrict__ Wcat, float* __restrict__ bcat) {
  int e = blockIdx.x * 256 + threadIdx.x;
  if (e < RPAD * C_DIM) {
    int r = e / C_DIM, c = e % C_DIM;
    float v = 0.0f;
    if (r < 128)        v = wmp[r * C_DIM + c];
    else if (r < 256)   v = wvp[(r - 128) * C_DIM + c];
    else if (r < RROWS) v = wm[(r - 256) * C_DIM + c];
    Wcat[e] = v;
  }
  if (e < RPAD) {
    float v = 0.0f;
    if (e < 128)        v = bmp[e];
    else if (e < 256)   v = bvp[e - 128];
    else if (e < RROWS) v = bm[e - 256];
    bcat[e] = v;
  }
}

// ---------------------------------------------------------------------------
// fp32 WMMA GEMM, workgroup tile 64(M) x 64(N), 8 waves arranged 4(M) x 2(N),
// each wave: 1 M-tile x 2 N-tiles (A-fragment reuse, 2 accumulators).
//   D[m][n] = sum_k A[m][k] * Bsrc(k,n)  (+ bias[m])
// B_IS_NK=true : Bsrc rows are n, contiguous k  (GEMM2: B = vproj[p][k])
// B_IS_NK=false: Bsrc rows are k, contiguous n  (GEMM1: B = x[k][n])
// Double-buffered LDS, staged with async b128 gathers, ASYNCcnt-synchronized.
// ---------------------------------------------------------------------------
template <bool B_IS_NK, bool HAS_BIAS>
__global__ void __launch_bounds__(256)
wmma_gemm_64x64(const float* __restrict__ Aa, int lda, long aStride,
                const float* __restrict__ Bb, int ldb, long bStride,
                float* __restrict__ Cc, int ldc, long cStride,
                const float* __restrict__ bias, int K, int Mvalid) {
  __shared__ __align__(16) float As[2][64 * LDA_S];
  __shared__ __align__(16) float Bsm[2][64 * LDA_S];  // 2304 >= KC*LDB_KN=2176

  const int t    = threadIdx.x;
  const int w    = t >> 5;
  const int lane = t & 31;
  const int half = lane >> 4;
  const int l    = lane & 15;
  const int mt   = w >> 1;   // 0..3
  const int ng   = w & 1;    // 0..1 (32-wide N group)
  const int wgM  = blockIdx.y * 64;
  const int wgN  = blockIdx.x * 64;

  const float* A = Aa + (long)blockIdx.z * aStride;
  const float* B = Bb + (long)blockIdx.z * bStride;
  float*       D = Cc + (long)blockIdx.z * cStride;

  const int arow = mt * 16 + l;
  const int bn0  = ng * 32 + l;

  v8f acc0 = {}, acc1 = {};

  auto issue = [&](int k0, int sel) {
    uint32_t aB = lds_off(&As[sel][0]);
    uint32_t bB = lds_off(&Bsm[sel][0]);
#pragma unroll
    for (int i = 0; i < 2; ++i) {           // A: 64 rows x 8 b128 chunks
      int e = t + i * 256;
      int r = e >> 3, k4 = (e & 7) * 4;
      async_ld_b128(aB + (uint32_t)(r * LDA_S + k4) * 4,
                    A + (long)(wgM + r) * lda + (k0 + k4));
    }
    if (B_IS_NK) {
#pragma unroll
      for (int i = 0; i < 2; ++i) {         // B[n][k]: 64 rows x 8 chunks
        int e = t + i * 256;
        int n = e >> 3, k4 = (e & 7) * 4;
        async_ld_b128(bB + (uint32_t)(n * LDA_S + k4) * 4,
                      B + (long)(wgN + n) * ldb + (k0 + k4));
      }
    } else {
#pragma unroll
      for (int i = 0; i < 2; ++i) {         // B[k][n]: 32 rows x 16 chunks
        int e = t + i * 256;
        int kk = e >> 4, n4 = (e & 15) * 4;
        async_ld_b128(bB + (uint32_t)(kk * LDB_KN + n4) * 4,
                      B + (long)(k0 + kk) * ldb + (wgN + n4));
      }
    }
  };

  auto compute = [&](int sel) {
#pragma unroll
    for (int kk = 0; kk < KC; kk += 4) {
      int kb = kk + half * 2;
      v2f a = *(const v2f*)&As[sel][arow * LDA_S + kb];
      v2f b0, b1;
      if (B_IS_NK) {
        b0 = *(const v2f*)&Bsm[sel][bn0 * LDA_S + kb];
        b1 = *(const v2f*)&Bsm[sel][(bn0 + 16) * LDA_S + kb];
      } else {
        b0.x = Bsm[sel][kb * LDB_KN + bn0];
        b0.y = Bsm[sel][(kb + 1) * LDB_KN + bn0];
        b1.x = Bsm[sel][kb * LDB_KN + bn0 + 16];
        b1.y = Bsm[sel][(kb + 1) * LDB_KN + bn0 + 16];
      }
      acc0 = wmma_f32(a, b0, acc0);
      acc1 = wmma_f32(a, b1, acc1);
    }
  };

  issue(0, 0);
  async_wait0();
  __syncthreads();
  const int nC = K / KC;
  for (int c = 0; c < nC; ++c) {
    int sel = c & 1;
    if (c + 1 < nC) issue((c + 1) * KC, sel ^ 1);  // overlap with compute
    compute(sel);
    async_wait0();
    __syncthreads();
  }

  // C/D layout: VGPR i -> (M=i, N=l) lanes 0-15 ; (M=i+8, N=l) lanes 16-31
#pragma unroll
  for (int j = 0; j < 2; ++j) {
    int cg = wgN + ng * 32 + j * 16 + l;
#pragma unroll
    for (int i = 0; i < 8; ++i) {
      int rg = wgM + mt * 16 + i + half * 8;
      if (rg < Mvalid) {
        float v = (j ? acc1[i] : acc0[i]);
        if (HAS_BIAS) v += bias[rg];
        D[(long)rg * ldc + cg] = v;
      }
    }
  }
}

// ---------------------------------------------------------------------------
// Row softmax over N=9216 (in place on the mask-logit rows of Y).
// ---------------------------------------------------------------------------
__global__ void __launch_bounds__(256)
softmax_rows(float* __restrict__ Yb) {
  const int m = blockIdx.x, b = blockIdx.y;
  float* row = Yb + ((long)b * RY + 256 + m) * N_DIM;
  __shared__ float red[256];
  const int t = threadIdx.x;

  float mx = -3.4e38f;
  for (int n = t; n < N_DIM; n += 256) mx = fmaxf(mx, row[n]);
  red[t] = mx; __syncthreads();
  for (int s = 128; s > 0; s >>= 1) {
    if (t < s) red[t] = fmaxf(red[t], red[t + s]);
    __syncthreads();
  }
  mx = red[0]; __syncthreads();

  float sum = 0.0f;
  for (int n = t; n < N_DIM; n += 256) sum += __expf(row[n] - mx);
  red[t] = sum; __syncthreads();
  for (int s = 128; s > 0; s >>= 1) {
    if (t < s) red[t] += red[t + s];
    __syncthreads();
  }
  const float inv = 1.0f / red[0];
  for (int n = t; n < N_DIM; n += 256) row[n] = __expf(row[n] - mx) * inv;
}

// ---------------------------------------------------------------------------
// cm2[b][m][p] = sum_n cmask[b][m][n] * mproj[b][p][n]   (K = 9216)
// One workgroup per batch, M padded 8->16 (rows 8-15 zeroed once), wave w
// owns P tile p0=16*w. Async-staged, double-buffered.
// ---------------------------------------------------------------------------
__global__ void __launch_bounds__(256)
wmma_gemm_cm2(const float* __restrict__ Yb, float* __restrict__ cm2) {
  const int b = blockIdx.x;
  const float* cm = Yb + ((long)b * RY + 256) * N_DIM;  // [8][N]
  const float* mp = Yb + (long)b * RY * N_DIM;          // mproj [128][N]

  __shared__ __align__(16) float Acm[2][16 * LDA_S];
  __shared__ __align__(16) float Bs[2][128 * LDA_S];

  const int t = threadIdx.x;
  const int w = t >> 5, lane = t & 31, half = lane >> 4, l = lane & 15;
  const int p0 = w * 16;
  v8f acc = {};

  // zero the padded A rows 8..15 once (async loads only touch rows 0..7)
  for (int i = t; i < 8 * LDA_S; i += 256) {
    Acm[0][8 * LDA_S + i] = 0.0f;
    Acm[1][8 * LDA_S + i] = 0.0f;
  }

  auto issue = [&](int k0, int sel) {
    uint32_t aB = lds_off(&Acm[sel][0]);
    uint32_t bB = lds_off(&Bs[sel][0]);
    if (t < 64) {                        // A: 8 rows x 8 b128 chunks
      int r = t >> 3, k4 = (t & 7) * 4;
      async_ld_b128(aB + (uint32_t)(r * LDA_S + k4) * 4,
                    cm + (long)r * N_DIM + k0 + k4);
    }
#pragma unroll
    for (int i = 0; i < 4; ++i) {        // B: 128 rows x 8 chunks
      int e = t + i * 256;
      int p = e >> 3, k4 = (e & 7) * 4;
      async_ld_b128(bB + (uint32_t)(p * LDA_S + k4) * 4,
                    mp + (long)p * N_DIM + k0 + k4);
    }
  };

  issue(0, 0);
  async_wait0();
  __syncthreads();
  const int nC = N_DIM / KC;
  for (int c = 0; c < nC; ++c) {
    int sel = c & 1;
    if (c + 1 < nC) issue((c + 1) * KC, sel ^ 1);
#pragma unroll
    for (int kk = 0; kk < KC; kk += 4) {
      int kb = kk + half * 2;
      v2f a  = *(const v2f*)&Acm[sel][l * LDA_S + kb];
      v2f bb = *(const v2f*)&Bs[sel][(p0 + l) * LDA_S + kb];
      acc = wmma_f32(a, bb, acc);
    }
    async_wait0();
    __syncthreads();
  }

  if (half == 0) {  // only rows m=0..7 valid
#pragma unroll
    for (int i = 0; i < 8; ++i)
      cm2[(long)b * M_DIM * P_DIM + i * P_DIM + p0 + l] = acc[i];
  }
}

// ---------------------------------------------------------------------------
// Per batch: softmax(cm2 over P); ctx2[m][c] = sum_p ctx[c][p]*cm2[m][p];
// h = w1 . ctx2_flat + b1 ; LayerNorm(PL) ; ReLU ; addv = w2 . h + b2.
// ---------------------------------------------------------------------------
__global__ void __launch_bounds__(256)
ctx_mlp(const float* __restrict__ cm2, const float* __restrict__ ctx,
        const float* __restrict__ w1, const float* __restrict__ b1,
        const float* __restrict__ g_ln, const float* __restrict__ beta_ln,
        const float* __restrict__ w2, const float* __restrict__ b2,
        float* __restrict__ addv) {
  const int b = blockIdx.x;
  __shared__ float cmS[M_DIM * P_DIM];
  __shared__ float ctx2f[M_DIM * C_DIM];
  __shared__ float part[256];
  __shared__ float hv[PL_DIM];
  __shared__ float stat[2];
  const int t = threadIdx.x;

  for (int i = t; i < M_DIM * P_DIM; i += 256)
    cmS[i] = cm2[(long)b * M_DIM * P_DIM + i];
  __syncthreads();

  if (t < M_DIM) {
    float* r = cmS + t * P_DIM;
    float mx = -3.4e38f;
    for (int p = 0; p < P_DIM; ++p) mx = fmaxf(mx, r[p]);
    float s = 0.0f;
    for (int p = 0; p < P_DIM; ++p) { float e = __expf(r[p] - mx); r[p] = e; s += e; }
    float inv = 1.0f / s;
    for (int p = 0; p < P_DIM; ++p) r[p] *= inv;
  }
  __syncthreads();

  const float* cb = ctx + (long)b * C_DIM * P_DIM;
  for (int j = 0; j < 16; ++j) {
    int i = t + j * 256;
    int m = i >> 9, c = i & 511;
    const float* cr = cb + (long)c * P_DIM;
    const float* mr = cmS + m * P_DIM;
    float a = 0.0f;
#pragma unroll 4
    for (int p = 0; p < P_DIM; ++p) a += cr[p] * mr[p];
    ctx2f[i] = a;  // flatten order m*C + c
  }
  __syncthreads();

  {
    int o = t >> 3, seg = t & 7;
    const float* wr = w1 + (long)o * (M_DIM * C_DIM) + seg * 512;
    const float* xr = ctx2f + seg * 512;
    float a = 0.0f;
    for (int j = 0; j < 512; ++j) a += wr[j] * xr[j];
    part[t] = a;
  }
  __syncthreads();
  if (t < PL_DIM) {
    float s = b1[t];
    for (int g = 0; g < 8; ++g) s += part[t * 8 + g];
    hv[t] = s;
  }
  __syncthreads();
  if (t == 0) {
    float mu = 0.0f;
    for (int i = 0; i < PL_DIM; ++i) mu += hv[i];
    mu *= (1.0f / PL_DIM);
    float var = 0.0f;
    for (int i = 0; i < PL_DIM; ++i) { float d = hv[i] - mu; var += d * d; }
    var *= (1.0f / PL_DIM);
    stat[0] = mu;
    stat[1] = rsqrtf(var + 1e-5f);
  }
  __syncthreads();
  if (t < PL_DIM) {
    float v = (hv[t] - stat[0]) * stat[1] * g_ln[t] + beta_ln[t];
    hv[t] = fmaxf(v, 0.0f);
  }
  __syncthreads();
  for (int c = t; c < C_DIM; c += 256) {
    float a = b2[c];
    const float* wr = w2 + (long)c * PL_DIM;
#pragma unroll
    for (int pl = 0; pl < PL_DIM; ++pl) a += wr[pl] * hv[pl];
    addv[(long)b * C_DIM + c] = a;
  }
}

// ---------------------------------------------------------------------------
// out[b][c][n] = x[b][c][n] + addv[b][c]   (float4, N % 4 == 0)
// ---------------------------------------------------------------------------
__global__ void __launch_bounds__(256)
add_bcast(const float* __restrict__ x, const float* __restrict__ addv,
          float* __restrict__ out) {
  long i4 = (long)blockIdx.x * 256 + threadIdx.x;
  long e  = i4 * 4;
  int bc  = (int)(e / N_DIM);  // == b*C + c
  float a = addv[bc];
  float4 v = ((const float4*)x)[i4];
  v.x += a; v.y += a; v.z += a; v.w += a;
  ((float4*)out)[i4] = v;
}

// ---------------------------------------------------------------------------
extern "C" void kernel_launch(void* const* d_in, const int* in_sizes, int n_in,
                              void* d_out, int out_size, void* d_ws, size_t ws_size,
                              hipStream_t stream) {
  (void)in_sizes; (void)n_in; (void)out_size; (void)ws_size;

  const float* x      = (const float*)d_in[0];
  const float* w_mask = (const float*)d_in[1];
  const float* b_mask = (const float*)d_in[2];
  const float* w_mp   = (const float*)d_in[3];
  const float* b_mp   = (const float*)d_in[4];
  const float* w_vp   = (const float*)d_in[5];
  const float* b_vp   = (const float*)d_in[6];
  const float* w1     = (const float*)d_in[7];
  const float* b1     = (const float*)d_in[8];
  const float* g_ln   = (const float*)d_in[9];
  const float* bln    = (const float*)d_in[10];
  const float* w2     = (const float*)d_in[11];
  const float* b2     = (const float*)d_in[12];
  float* out = (float*)d_out;
  float* ws  = (float*)d_ws;

  float* Wcat = ws + OFF_WCAT;
  float* bcat = ws + OFF_BCAT;
  float* Y    = ws + OFF_Y;                           // [B][RY][N]
  float* cm2  = Y + (size_t)B_DIM * RY * N_DIM;       // [B][M][P]
  float* ctx  = cm2 + (size_t)B_DIM * M_DIM * P_DIM;  // [B][C][P]
  float* addv = ctx + (size_t)B_DIM * C_DIM * P_DIM;  // [B][C]

  // 0) pack weights/biases
  prep_kernel<<<dim3((RPAD * C_DIM + 255) / 256), 256, 0, stream>>>(
      w_mask, b_mask, w_mp, b_mp, w_vp, b_vp, Wcat, bcat);

  // 1) fused projections: Y[b][r][n] = Wcat[r][:] . x[b][:][n] + bcat[r]
  wmma_gemm_64x64<false, true><<<dim3(N_DIM / 64, RPAD / 64, B_DIM), 256, 0, stream>>>(
      Wcat, C_DIM, 0L,
      x, N_DIM, (long)C_DIM * N_DIM,
      Y, N_DIM, (long)RY * N_DIM,
      bcat, C_DIM, RY);

  // 2) softmax over N on mask logits (rows 256..263 of each batch panel)
  softmax_rows<<<dim3(M_DIM, B_DIM), 256, 0, stream>>>(Y);

  // 3) cm2[b][m][p] = cmask . mproj^T  (K = 9216)
  wmma_gemm_cm2<<<dim3(B_DIM), 256, 0, stream>>>(Y, cm2);

  // 4) ctx[b][c][p] = x[b][c][:] . vproj[b][p][:]  (K = 9216)
  wmma_gemm_64x64<true, false><<<dim3(P_DIM / 64, C_DIM / 64, B_DIM), 256, 0, stream>>>(
      x, N_DIM, (long)C_DIM * N_DIM,
      Y + (size_t)128 * N_DIM, N_DIM, (long)RY * N_DIM,  // vproj rows of Y
      ctx, P_DIM, (long)C_DIM * P_DIM,
      nullptr, N_DIM, C_DIM);

  // 5) softmax(P) + ctx.cmask^T + MLP (w1 -> LN -> ReLU -> w2) -> addv[b][c]
  ctx_mlp<<<dim3(B_DIM), 256, 0, stream>>>(cm2, ctx, w1, b1, g_ln, bln, w2, b2, addv);

  // 6) broadcast add epilogue
  add_bcast<<<dim3((int)(((long)B_DIM * C_DIM * N_DIM / 4 + 255) / 256)), 256, 0, stream>>>(
      x, addv, out);
}